// CatSimLoss_59158879535450
// MI455X (gfx1250) — compile-verified
//
#include <hip/hip_runtime.h>
#include <math.h>

#define EPSV 1e-8f
#define NB 4          // batches
#define NP 6144       // points per batch
#define NK 8          // keypoints
#define NV 42         // cage vertices
#define NF 80         // cage faces
#define THRESH2 0.16f // 0.4^2

typedef float v2f __attribute__((ext_vector_type(2)));
typedef float v8f __attribute__((ext_vector_type(8)));

// ---------------------------------------------------------------- phase 1
// One block per (batch, cage-vertex). 256 threads hold all 6144 points in
// registers; 100 serial shrink iterations, per-iter min via shfl + LDS.
__global__ void cage_opt_kernel(const float* __restrict__ catModel,
                                const float* __restrict__ tmpl,
                                float* __restrict__ cage)
{
    int b = blockIdx.x / NV;
    int v = blockIdx.x % NV;
    int tid = threadIdx.x;
    const float* pc = catModel + (size_t)b * NP * 3;

    float px[24], py[24], pz[24];
#pragma unroll
    for (int k = 0; k < 24; ++k) {
        int p = tid + (k << 8);
        px[k] = pc[p * 3 + 0];
        py[k] = pc[p * 3 + 1];
        pz[k] = pc[p * 3 + 2];
    }
    // template_vertices is (1,3,42)
    float cx = tmpl[0 * NV + v], cy = tmpl[1 * NV + v], cz = tmpl[2 * NV + v];

    __shared__ float smem[9];
    for (int it = 0; it < 100; ++it) {
        float m = __int_as_float(0x7F800000);
#pragma unroll
        for (int k = 0; k < 24; ++k) {
            float dx = px[k] - cx, dy = py[k] - cy, dz = pz[k] - cz;
            float d2 = fmaf(dx, dx, fmaf(dy, dy, dz * dz));
            m = fminf(m, d2);
        }
#pragma unroll
        for (int o = 16; o >= 1; o >>= 1)
            m = fminf(m, __shfl_xor(m, o, 32));
        if ((tid & 31) == 0) smem[tid >> 5] = m;
        __syncthreads();
        if (tid == 0) {
            float mm = smem[0];
#pragma unroll
            for (int w = 1; w < 8; ++w) mm = fminf(mm, smem[w]);
            smem[8] = mm;
        }
        __syncthreads();
        float mind2 = smem[8];
        if (mind2 > THRESH2) { cx *= 0.99f; cy *= 0.99f; cz *= 0.99f; }
    }
    if (tid == 0) {
        float* o = cage + (size_t)(b * NV + v) * 3;
        o[0] = cx; o[1] = cy; o[2] = cz;
    }
}

// ---------------------------------------------------------------- phase 2
__global__ void new_cage_kernel(const float* __restrict__ catKeys,
                                const float* __restrict__ insKeys,
                                const float* __restrict__ influ,
                                const float* __restrict__ cage,
                                float* __restrict__ ncage)
{
    int i = blockIdx.x * blockDim.x + threadIdx.x; // b*NV + v
    if (i >= NB * NV) return;
    int b = i / NV, v = i % NV;
    float ox = 0.f, oy = 0.f, oz = 0.f;
#pragma unroll
    for (int k = 0; k < NK; ++k) {
        float w = influ[(b * NK + k) * NV + v];
        const float* ik = insKeys + (size_t)(b * NK + k) * 3;
        const float* ck = catKeys + (size_t)(b * NK + k) * 3;
        ox = fmaf(ik[0] - ck[0], w, ox);
        oy = fmaf(ik[1] - ck[1], w, oy);
        oz = fmaf(ik[2] - ck[2], w, oz);
    }
    const float* c = cage + (size_t)i * 3;
    float* o = ncage + (size_t)i * 3;
    o[0] = c[0] + ox; o[1] = c[1] + oy; o[2] = c[2] + oz;
}

// ---------------------------------------------------------------- phase 3
// Mean-value-coordinate deform, fully fused (no (B,N,42) weight matrix).
__global__ void mvc_deform_kernel(const float* __restrict__ catModel,
                                  const float* __restrict__ cage,
                                  const float* __restrict__ ncage,
                                  const int* __restrict__ faces,
                                  float* __restrict__ deformed)
{
    __shared__ float sC[NV * 3];
    __shared__ float sN[NV * 3];
    __shared__ int   sF[NF * 3];
    const int bpb = NP / 256;
    int b = blockIdx.x / bpb;
    int p = (blockIdx.x % bpb) * 256 + threadIdx.x;
    if (threadIdx.x < NV * 3) {
        sC[threadIdx.x] = cage[b * NV * 3 + threadIdx.x];
        sN[threadIdx.x] = ncage[b * NV * 3 + threadIdx.x];
    }
    if (threadIdx.x < NF * 3) sF[threadIdx.x] = faces[threadIdx.x];
    __syncthreads();

    const float* q = catModel + (size_t)(b * NP + p) * 3;
    float qx = q[0], qy = q[1], qz = q[2];
    float ax = 0.f, ay = 0.f, az = 0.f, sw = 0.f;

    for (int f = 0; f < NF; ++f) {
        int idx0 = sF[f * 3 + 0], idx1 = sF[f * 3 + 1], idx2 = sF[f * 3 + 2];
        int idx[3] = { idx0, idx1, idx2 };
        float ux[3], uy[3], uz[3], dd[3];
#pragma unroll
        for (int k = 0; k < 3; ++k) {
            const float* vv = &sC[idx[k] * 3];
            float ex = vv[0] - qx, ey = vv[1] - qy, ez = vv[2] - qz;
            float d = sqrtf(fmaf(ex, ex, fmaf(ey, ey, ez * ez)));
            dd[k] = d;
            float inv = 1.0f / fmaxf(d, EPSV);
            ux[k] = ex * inv; uy[k] = ey * inv; uz[k] = ez * inv;
        }
        float th[3], st[3];
#pragma unroll
        for (int j = 0; j < 3; ++j) {
            int jp = (j + 1) % 3, jm = (j + 2) % 3;
            float lx = ux[jp] - ux[jm], ly = uy[jp] - uy[jm], lz = uz[jp] - uz[jm];
            float l = sqrtf(fmaf(lx, lx, fmaf(ly, ly, lz * lz)));
            float hl = fminf(fmaxf(0.5f * l, 0.f), 1.f);
            th[j] = 2.f * asinf(hl);
            st[j] = sinf(th[j]);
        }
        float h  = 0.5f * (th[0] + th[1] + th[2]);
        float sh = sinf(h);
        float cxx = uy[1] * uz[2] - uz[1] * uy[2];
        float cyy = uz[1] * ux[2] - ux[1] * uz[2];
        float czz = ux[1] * uy[2] - uy[1] * ux[2];
        float det = ux[0] * cxx + uy[0] * cyy + uz[0] * czz;
        float sgn = (det > 0.f) ? 1.f : ((det < 0.f) ? -1.f : 0.f);
        float cc[3], ss[3];
#pragma unroll
        for (int j = 0; j < 3; ++j) {
            int jp = (j + 1) % 3, jm = (j + 2) % 3;
            float cj = 2.f * sh * sinf(h - th[j]) /
                       fmaxf(st[jp] * st[jm], EPSV) - 1.f;
            cc[j] = cj;
            ss[j] = sgn * sqrtf(fmaxf(1.f - cj * cj, 0.f));
        }
        bool valid = (fabsf(ss[0]) > EPSV) && (fabsf(ss[1]) > EPSV) &&
                     (fabsf(ss[2]) > EPSV);
#pragma unroll
        for (int j = 0; j < 3; ++j) {
            int jp = (j + 1) % 3, jm = (j + 2) % 3;
            float num = th[j] - cc[jp] * th[jm] - cc[jm] * th[jp];
            float den = dd[j] * st[jp] * ss[jm];
            den = (fabsf(den) > EPSV) ? den : ((den >= 0.f) ? EPSV : -EPSV);
            float w = valid ? (num / den) : 0.f;
            const float* nv = &sN[idx[j] * 3];
            ax = fmaf(w, nv[0], ax);
            ay = fmaf(w, nv[1], ay);
            az = fmaf(w, nv[2], az);
            sw += w;
        }
    }
    float swc = (fabsf(sw) > EPSV) ? sw : EPSV;
    float inv = 1.0f / swc;
    float* o = deformed + (size_t)(b * NP + p) * 3;
    o[0] = ax * inv; o[1] = ay * inv; o[2] = az * inv;
}

// ---------------------------------------------------------------- phase 4
__global__ void init_inf_kernel(float* __restrict__ buf, int n)
{
    int i = blockIdx.x * blockDim.x + threadIdx.x;
    if (i < n) buf[i] = __int_as_float(0x7F800000);
}

// Pairwise min-squared-distance via V_WMMA_F32_16X16X4_F32.
// d2 = |x|^2 + |y|^2 - 2 x.y ; the Gram term is a K=3 (padded to 4) GEMM.
// Each wave owns TWO 16-row A tiles (32 rows) and the j loop is unrolled x2:
// per unrolled iter -> 2 y loads, 4 WMMAs, 2 column atomics.
__global__ void dist_min_kernel(const float* __restrict__ Xall, // deformed
                                const float* __restrict__ Yall, // insModel
                                float* __restrict__ rowMin,
                                float* __restrict__ colMin)
{
    int b    = blockIdx.x;
    int lane = threadIdx.x & 31;
    int wave = threadIdx.x >> 5;
    int half = lane >> 4;
    int lq   = lane & 15;
    int i0   = (blockIdx.y * 8 + wave) * 32;   // 32 rows per wave

    const float* X = Xall + (size_t)b * NP * 3;
    const float* Y = Yall + (size_t)b * NP * 3;

    // A operands: two 16-row tiles
    int ip0 = i0 + lq;
    int ip1 = i0 + 16 + lq;
    float a00 = X[ip0 * 3 + 0], a01 = X[ip0 * 3 + 1], a02 = X[ip0 * 3 + 2];
    float a10 = X[ip1 * 3 + 0], a11 = X[ip1 * 3 + 1], a12 = X[ip1 * 3 + 2];
    float na0 = fmaf(a00, a00, fmaf(a01, a01, a02 * a02));
    float na1 = fmaf(a10, a10, fmaf(a11, a11, a12 * a12));
    v2f A0, A1;
    A0.x = half ? a02 : a00;  A0.y = half ? 0.f : a01;
    A1.x = half ? a12 : a10;  A1.y = half ? 0.f : a11;

    float rowN0[8], rowN1[8];
#pragma unroll
    for (int r = 0; r < 8; ++r) {
        rowN0[r] = __shfl(na0, r + 8 * half, 32);
        rowN1[r] = __shfl(na1, r + 8 * half, 32);
    }

    const float FINF = __int_as_float(0x7F800000);
    float rmin0[8], rmin1[8];
#pragma unroll
    for (int r = 0; r < 8; ++r) { rmin0[r] = FINF; rmin1[r] = FINF; }

    unsigned int* colM = (unsigned int*)(colMin + (size_t)b * NP);

    for (int jt = 0; jt < NP / 16; jt += 2) {
        int jp0 = jt * 16 + lq;
        int jp1 = jp0 + 16;
        // issue both y loads back-to-back (latency overlap)
        float b00 = Y[jp0 * 3 + 0], b01 = Y[jp0 * 3 + 1], b02 = Y[jp0 * 3 + 2];
        float b10 = Y[jp1 * 3 + 0], b11 = Y[jp1 * 3 + 1], b12 = Y[jp1 * 3 + 2];
        float nb0 = fmaf(b00, b00, fmaf(b01, b01, b02 * b02));
        float nb1 = fmaf(b10, b10, fmaf(b11, b11, b12 * b12));
        v2f B0, B1;
        B0.x = half ? b02 : b00;  B0.y = half ? 0.f : b01;
        B1.x = half ? b12 : b10;  B1.y = half ? 0.f : b11;

        v8f Z = {0.f, 0.f, 0.f, 0.f, 0.f, 0.f, 0.f, 0.f};
        v8f C00 = __builtin_amdgcn_wmma_f32_16x16x4_f32(
                      false, A0, false, B0, (short)0, Z, false, false);
        v8f C10 = __builtin_amdgcn_wmma_f32_16x16x4_f32(
                      false, A1, false, B0, (short)0, Z, false, false);
        v8f C01 = __builtin_amdgcn_wmma_f32_16x16x4_f32(
                      false, A0, false, B1, (short)0, Z, false, false);
        v8f C11 = __builtin_amdgcn_wmma_f32_16x16x4_f32(
                      false, A1, false, B1, (short)0, Z, false, false);

        float cmin0 = FINF, cmin1 = FINF;
#pragma unroll
        for (int r = 0; r < 8; ++r) {
            float d00 = fmaxf(fmaf(-2.f, C00[r], rowN0[r] + nb0), 0.f);
            float d10 = fmaxf(fmaf(-2.f, C10[r], rowN1[r] + nb0), 0.f);
            float d01 = fmaxf(fmaf(-2.f, C01[r], rowN0[r] + nb1), 0.f);
            float d11 = fmaxf(fmaf(-2.f, C11[r], rowN1[r] + nb1), 0.f);
            rmin0[r] = fminf(rmin0[r], fminf(d00, d01));
            rmin1[r] = fminf(rmin1[r], fminf(d10, d11));
            cmin0 = fminf(cmin0, fminf(d00, d10));
            cmin1 = fminf(cmin1, fminf(d01, d11));
        }
        cmin0 = fminf(cmin0, __shfl_xor(cmin0, 16, 32));
        cmin1 = fminf(cmin1, __shfl_xor(cmin1, 16, 32));
        if (half == 0) {
            atomicMin(&colM[jp0], __float_as_uint(cmin0));
            atomicMin(&colM[jp1], __float_as_uint(cmin1));
        }
    }

#pragma unroll
    for (int r = 0; r < 8; ++r) {
#pragma unroll
        for (int o = 1; o < 16; o <<= 1) {
            rmin0[r] = fminf(rmin0[r], __shfl_xor(rmin0[r], o, 32));
            rmin1[r] = fminf(rmin1[r], __shfl_xor(rmin1[r], o, 32));
        }
    }
    if (lq == 0) {
#pragma unroll
        for (int r = 0; r < 8; ++r) {
            rowMin[(size_t)b * NP + i0 + half * 8 + r]      = rmin0[r];
            rowMin[(size_t)b * NP + i0 + 16 + half * 8 + r] = rmin1[r];
        }
    }
}

// rowMin and colMin are contiguous; both means divide by B*NP.
__global__ void loss_kernel(const float* __restrict__ mins,
                            float* __restrict__ out)
{
    __shared__ float red[256];
    float s = 0.f;
    for (int i = threadIdx.x; i < 2 * NB * NP; i += 256) s += mins[i];
    red[threadIdx.x] = s;
    __syncthreads();
    for (int o = 128; o >= 1; o >>= 1) {
        if (threadIdx.x < o) red[threadIdx.x] += red[threadIdx.x + o];
        __syncthreads();
    }
    if (threadIdx.x == 0) out[0] = red[0] / (float)(NB * NP);
}

// ----------------------------------------------------------------
extern "C" void kernel_launch(void* const* d_in, const int* in_sizes, int n_in,
                              void* d_out, int out_size, void* d_ws, size_t ws_size,
                              hipStream_t stream)
{
    const float* catModel = (const float*)d_in[0];
    const float* catKeys  = (const float*)d_in[1];
    const float* catInflu = (const float*)d_in[2];
    const float* insKeys  = (const float*)d_in[3];
    const float* insModel = (const float*)d_in[4];
    const float* tmpl     = (const float*)d_in[5];
    const int*   faces    = (const int*)d_in[6];

    float* out = (float*)d_out;
    float* ws  = (float*)d_ws;

    float* cage   = ws;                       // B*42*3
    float* ncage  = cage + NB * NV * 3;       // B*42*3
    float* rowMin = ncage + NB * NV * 3;      // B*NP
    float* colMin = rowMin + NB * NP;         // B*NP

    const size_t modelElems = (size_t)NB * NP * 3;
    float* deformed = out + 2 * modelElems;
    float* loss     = out + 3 * modelElems;

    // Pass-through outputs 0 and 1.
    hipMemcpyAsync(out, catModel, modelElems * sizeof(float),
                   hipMemcpyDeviceToDevice, stream);
    hipMemcpyAsync(out + modelElems, insModel, modelElems * sizeof(float),
                   hipMemcpyDeviceToDevice, stream);

    cage_opt_kernel<<<NB * NV, 256, 0, stream>>>(catModel, tmpl, cage);
    new_cage_kernel<<<1, 256, 0, stream>>>(catKeys, insKeys, catInflu,
                                           cage, ncage);
    mvc_deform_kernel<<<NB * NP / 256, 256, 0, stream>>>(catModel, cage, ncage,
                                                         faces, deformed);
    init_inf_kernel<<<(2 * NB * NP + 255) / 256, 256, 0, stream>>>(rowMin,
                                                                   2 * NB * NP);
    dim3 g(NB, NP / 256);   // 8 waves/block, 32 rows/wave
    dist_min_kernel<<<g, 256, 0, stream>>>(deformed, insModel, rowMin, colMin);
    loss_kernel<<<1, 256, 0, stream>>>(rowMin, loss);
}